// IntraGNN_47210280517971
// MI455X (gfx1250) — compile-verified
//
#include <hip/hip_runtime.h>
#include <hip/hip_bf16.h>

#define Nn 8192   // total nodes
#define Mm 4096   // batch rows
#define Dd 128    // feature dim
#define Hh 128    // hidden dim

typedef __attribute__((ext_vector_type(2))) float v2f;
typedef __attribute__((ext_vector_type(8))) float v8f;

// ---------------------------------------------------------------------------
// Kernel 0: zero the global accumulators (ws is poisoned by the harness)
// ---------------------------------------------------------------------------
__global__ void init_accum_kernel(float* __restrict__ accum) {
    accum[0] = 0.0f;   // sum of kept adjacency values
    accum[1] = 0.0f;   // sum of per-row k
}

// ---------------------------------------------------------------------------
// Kernel 1: per-row exact top-k threshold via 4-pass radix select in LDS.
// One workgroup per batch row; row (32KB) staged in LDS once.
// ---------------------------------------------------------------------------
__global__ void select_kernel(const float* __restrict__ weights,
                              const int*   __restrict__ batch_idx,
                              const float* __restrict__ thr_ptr,
                              float* __restrict__ thr_row,
                              float* __restrict__ invmax_row,
                              float* __restrict__ accum) {
    __shared__ float    rowbuf[Nn];   // 32 KB
    __shared__ int      hist[256];
    __shared__ float    red[256];
    __shared__ unsigned s_prefix;
    __shared__ int      s_krem;

    const int m = blockIdx.x;
    const int t = threadIdx.x;
    const int src = batch_idx[m];
    const float* wrow = weights + (size_t)src * Nn;

    // ---- stage row into LDS (float4, coalesced) ----
    for (int i = 0; i < Nn / (256 * 4); ++i) {
        ((float4*)rowbuf)[t + i * 256] = ((const float4*)wrow)[t + i * 256];
    }
    __syncthreads();

    // ---- count positives ----
    float cnt = 0.0f, mx = 0.0f;
    for (int i = t; i < Nn; i += 256) {
        float v = rowbuf[i];
        if (v > 0.0f) { cnt += 1.0f; mx = fmaxf(mx, v); }
    }
    red[t] = cnt; __syncthreads();
    for (int s = 128; s > 0; s >>= 1) { if (t < s) red[t] += red[t + s]; __syncthreads(); }
    const float cntpos = red[0];
    __syncthreads();

    // ---- row max ----
    red[t] = mx; __syncthreads();
    for (int s = 128; s > 0; s >>= 1) { if (t < s) red[t] = fmaxf(red[t], red[t + s]); __syncthreads(); }
    const float rowmax = red[0];
    __syncthreads();

    const float thr = *thr_ptr;
    const int k = (int)ceilf(cntpos * thr);

    // ---- exact k-th largest via radix select on positive float bit patterns ----
    unsigned prefix = 0;
    int krem = k;
    if (k > 0) {
        for (int byte = 3; byte >= 0; --byte) {
            const int shift = byte * 8;
            hist[t] = 0;
            __syncthreads();
            const unsigned hmask = (byte == 3) ? 0u : (0xFFFFFFFFu << (shift + 8));
            for (int i = t; i < Nn; i += 256) {
                float v = rowbuf[i];
                if (v > 0.0f) {
                    unsigned u = __float_as_uint(v);
                    if ((u & hmask) == prefix)
                        atomicAdd(&hist[(u >> shift) & 255], 1);
                }
            }
            __syncthreads();
            if (t == 0) {
                int cum = 0, kr = krem, newkr = kr;
                unsigned bsel = 0;
                for (int b = 255; b >= 0; --b) {
                    int c = hist[b];
                    if (cum + c >= kr) { bsel = (unsigned)b; newkr = kr - cum; break; }
                    cum += c;
                }
                s_prefix = prefix | (bsel << shift);
                s_krem = newkr;
            }
            __syncthreads();
            prefix = s_prefix;
            krem = s_krem;
            __syncthreads();
        }
    }
    const float tsel = (k > 0) ? __uint_as_float(prefix) : __builtin_inff();

    // ---- sum of kept values ----
    float s = 0.0f;
    for (int i = t; i < Nn; i += 256) {
        float v = rowbuf[i];
        if (v >= tsel) s += v;
    }
    red[t] = s; __syncthreads();
    for (int st = 128; st > 0; st >>= 1) { if (t < st) red[t] += red[t + st]; __syncthreads(); }
    if (t == 0) {
        atomicAdd(&accum[0], red[0]);
        atomicAdd(&accum[1], (float)k);
        thr_row[m]    = tsel;
        invmax_row[m] = (rowmax > 0.0f) ? 1.0f / rowmax : 0.0f;
    }
}

// ---------------------------------------------------------------------------
// WMMA f32 16x16x4 tile MAC: wave computes 16x16 C tile, fragments per
// ISA 7.12.2 (32-bit A 16x4: lanes 0-15 K=0,1; lanes 16-31 K=2,3;
// B 4x16: V0 rows K=0/K=2, V1 rows K=1/K=3 across lane halves).
// ---------------------------------------------------------------------------
template <int APITCH, int BPITCH>
__device__ __forceinline__ void wave_mac_16x16x64(
        const float* __restrict__ At,   // [16][APITCH]
        const float* __restrict__ Bt,   // [64][BPITCH]
        int n0, int l15, int half, v8f& acc) {
    for (int kk = 0; kk < 64; kk += 4) {
#if __has_builtin(__builtin_amdgcn_wmma_f32_16x16x4_f32)
        v2f a, b;
        a.x = At[l15 * APITCH + kk + 2 * half + 0];
        a.y = At[l15 * APITCH + kk + 2 * half + 1];
        b.x = Bt[(kk + 2 * half + 0) * BPITCH + n0 + l15];
        b.y = Bt[(kk + 2 * half + 1) * BPITCH + n0 + l15];
        acc = __builtin_amdgcn_wmma_f32_16x16x4_f32(
            false, a, false, b, (short)0, acc, false, false);
#else
        // exact VALU fallback (same math, ignores fragment striping)
        for (int i = 0; i < 8; ++i) {
            int row = i + half * 8;
            float sv = acc[i];
            for (int j = 0; j < 4; ++j)
                sv = fmaf(At[row * APITCH + kk + j], Bt[(kk + j) * BPITCH + n0 + l15], sv);
            acc[i] = sv;
        }
#endif
    }
}

// ---------------------------------------------------------------------------
// Kernel 2: C1 = adj_norm @ features   (4096x8192)*(8192x128), fused mask.
// 256 WGs, 256 threads (8 waves). Each WG: 16 rows x 128 cols.
// ---------------------------------------------------------------------------
__global__ void gemm1_kernel(const float* __restrict__ weights,
                             const float* __restrict__ features,
                             const int*   __restrict__ batch_idx,
                             const float* __restrict__ thr_row,
                             const float* __restrict__ invmax_row,
                             float* __restrict__ C1) {
    __shared__ float At[16][68];    // 16 x 64 (+4 pad)
    __shared__ float Ft[64][136];   // 64 x 128 (+8 pad)
    __shared__ float thrS[16], invS[16];
    __shared__ int   srcS[16];

    const int m0 = blockIdx.x * 16;
    const int t  = threadIdx.x;
    const int lane = t & 31, wv = t >> 5;
    const int half = lane >> 4, l15 = lane & 15;
    const int n0 = wv * 16;

    if (t < 16) {
        srcS[t] = batch_idx[m0 + t];
        thrS[t] = thr_row[m0 + t];
        invS[t] = invmax_row[m0 + t];
    }
    __syncthreads();

    v8f acc = {};
    for (int k0 = 0; k0 < Nn; k0 += 64) {
        // A tile: mask + scale at load.  thread -> row t>>4, 4 cols
        {
            const int r = t >> 4, c = (t & 15) * 4;
            const float4 v = *(const float4*)(weights + (size_t)srcS[r] * Nn + k0 + c);
            const float th = thrS[r], im = invS[r];
            At[r][c + 0] = (v.x >= th) ? v.x * im : 0.0f;
            At[r][c + 1] = (v.y >= th) ? v.y * im : 0.0f;
            At[r][c + 2] = (v.z >= th) ? v.z * im : 0.0f;
            At[r][c + 3] = (v.w >= th) ? v.w * im : 0.0f;
        }
        // F tile: 64x128, thread -> row t>>2, 32 cols
        {
            const int r = t >> 2, cs = (t & 3) * 32;
            const float4* frow = (const float4*)(features + (size_t)(k0 + r) * Dd + cs);
            for (int j = 0; j < 8; ++j) {
                float4 v = frow[j];
                int c = cs + j * 4;
                Ft[r][c] = v.x; Ft[r][c + 1] = v.y; Ft[r][c + 2] = v.z; Ft[r][c + 3] = v.w;
            }
        }
        __syncthreads();
        wave_mac_16x16x64<68, 136>(&At[0][0], &Ft[0][0], n0, l15, half, acc);
        __syncthreads();
    }
    for (int i = 0; i < 8; ++i) {
        int row = m0 + i + half * 8;
        C1[(size_t)row * Dd + n0 + l15] = acc[i];
    }
}

// ---------------------------------------------------------------------------
// Kernel 3: out = leakyrelu(C1 @ w_gnn)   (4096x128)*(128x128)
// ---------------------------------------------------------------------------
__global__ void gemm2_kernel(const float* __restrict__ C1,
                             const float* __restrict__ wg,
                             const float* __restrict__ slope_ptr,
                             float* __restrict__ out) {
    __shared__ float At[16][68];
    __shared__ float Bt[64][136];

    const int m0 = blockIdx.x * 16;
    const int t  = threadIdx.x;
    const int lane = t & 31, wv = t >> 5;
    const int half = lane >> 4, l15 = lane & 15;
    const int n0 = wv * 16;
    const float slope = *slope_ptr;

    v8f acc = {};
    for (int k0 = 0; k0 < Dd; k0 += 64) {
        {
            const int r = t >> 4, c = (t & 15) * 4;
            float4 v = *(const float4*)(C1 + (size_t)(m0 + r) * Dd + k0 + c);
            At[r][c + 0] = v.x; At[r][c + 1] = v.y; At[r][c + 2] = v.z; At[r][c + 3] = v.w;
        }
        {
            const int r = t >> 2, cs = (t & 3) * 32;
            const float4* brow = (const float4*)(wg + (size_t)(k0 + r) * Hh + cs);
            for (int j = 0; j < 8; ++j) {
                float4 v = brow[j];
                int c = cs + j * 4;
                Bt[r][c] = v.x; Bt[r][c + 1] = v.y; Bt[r][c + 2] = v.z; Bt[r][c + 3] = v.w;
            }
        }
        __syncthreads();
        wave_mac_16x16x64<68, 136>(&At[0][0], &Bt[0][0], n0, l15, half, acc);
        __syncthreads();
    }
    for (int i = 0; i < 8; ++i) {
        int row = m0 + i + half * 8;
        float v = acc[i];
        v = (v >= 0.0f) ? v : slope * v;
        out[(size_t)row * Hh + n0 + l15] = v;
    }
}

// ---------------------------------------------------------------------------
// Kernel 4: view_score
// ---------------------------------------------------------------------------
__global__ void score_kernel(const float* __restrict__ accum,
                             float* __restrict__ out) {
    out[(size_t)Mm * Hh] = accum[0] / accum[1];
}

// ---------------------------------------------------------------------------
extern "C" void kernel_launch(void* const* d_in, const int* in_sizes, int n_in,
                              void* d_out, int out_size, void* d_ws, size_t ws_size,
                              hipStream_t stream) {
    const float* features = (const float*)d_in[0];   // (N, D)
    const float* weights  = (const float*)d_in[1];   // (N, N)
    const float* w_gnn    = (const float*)d_in[2];   // (D, H)
    // d_in[3] edge_feats: unused
    const float* rl_thr   = (const float*)d_in[4];   // scalar
    const int*   batch    = (const int*)  d_in[5];   // (M,)
    const float* slope    = (const float*)d_in[6];   // scalar

    float* out = (float*)d_out;

    float* W          = (float*)d_ws;
    float* thr_row    = W;              // 4096
    float* invmax_row = W + 4096;       // 4096
    float* accum      = W + 8192;       // 2
    float* C1         = W + 16384;      // 4096*128

    init_accum_kernel<<<1, 1, 0, stream>>>(accum);
    select_kernel<<<Mm, 256, 0, stream>>>(weights, batch, rl_thr,
                                          thr_row, invmax_row, accum);
    gemm1_kernel<<<Mm / 16, 256, 0, stream>>>(weights, features, batch,
                                              thr_row, invmax_row, C1);
    gemm2_kernel<<<Mm / 16, 256, 0, stream>>>(C1, w_gnn, slope, out);
    score_kernel<<<1, 1, 0, stream>>>(accum, out);
}